// GroupDecoder_31645319037327
// MI455X (gfx1250) — compile-verified
//
#include <hip/hip_runtime.h>
#include <hip/hip_bf16.h>

typedef __attribute__((ext_vector_type(2))) float v2f;
typedef __attribute__((ext_vector_type(8))) float v8f;

#define B_  128
#define S_  12
#define L_  128
#define E2_ 1024
#define H_  512

#define TK_   16   // K chunk staged in LDS
#define LDS_S 20   // padded LDS row stride (banks 20*m mod 64 all distinct, 16B aligned)

// ---------------------------------------------------------------------------
// WMMA f32 GEMM:  D[M x N] = A[M x K] * W^T (+ bias[n])
// A row-major (lda = K). W is [N x K] row-major with row stride ldw (supports
// column-sliced weights like W_p1[:, :H] / W_p1[:, H:] via base-ptr offset).
//
// Block = 128 threads = 4 waves, computes a 64x64 output tile.
// Wave w owns rows 16w..16w+15 and all 64 columns (4 x v8f accumulators).
// A and W K-slices (64x16 each) are double-buffered in LDS; the W slice is
// shared by all 4 waves (4x reuse), and global b128 loads for chunk i+1 are
// issued before the 16 WMMAs of chunk i, so HBM latency overlaps matrix ops
// instead of forcing s_wait_loadcnt 0 before every v_wmma.
//
// V_WMMA_F32_16X16X4_F32 operand layout:
//   A 16x4:  lanes 0-15 -> M=lane, K={0,1}; lanes 16-31 -> K={2,3}
//   B 4x16:  lanes 0-15 -> N=lane, K={0,1}; lanes 16-31 -> K={2,3}
//   C/D:     VGPR v -> row base+v (+8 for lanes>=16), col = lane&15
// ---------------------------------------------------------------------------
__global__ __launch_bounds__(128) void wmma_gemm_nt(
    const float* __restrict__ A, const float* __restrict__ W,
    const float* __restrict__ bias, float* __restrict__ D,
    int M, int N, int K, int ldw)
{
    __shared__ float As[2][64 * LDS_S];
    __shared__ float Ws[2][64 * LDS_S];

    const int tid   = threadIdx.x;        // 0..127
    const int wave  = tid >> 5;           // 0..3
    const int lane  = tid & 31;
    const int lm    = lane & 15;
    const int khalf = (lane >> 4) << 1;   // 0 for lanes 0-15, 2 for 16-31

    const int mt = blockIdx.x;            // 64-row tile
    const int nt = blockIdx.y;            // 64-col tile

    // Staging: thread t fills row (t>>1), k-offset (t&1)*8 -> 8 floats each.
    const int lr = tid >> 1;
    const int lk = (tid & 1) * 8;
    const float* __restrict__ Ag = A + (size_t)(mt * 64 + lr) * K + lk;
    const float* __restrict__ Wg = W + (size_t)(nt * 64 + lr) * ldw + lk;
    float* const asd = &As[0][lr * LDS_S + lk];
    float* const wsd = &Ws[0][lr * LDS_S + lk];
    const int bufstride = 64 * LDS_S;

    auto stage = [&](int buf, int k0) {
        const float4 a0 = *(const float4*)(Ag + k0);
        const float4 a1 = *(const float4*)(Ag + k0 + 4);
        const float4 w0 = *(const float4*)(Wg + k0);
        const float4 w1 = *(const float4*)(Wg + k0 + 4);
        float* as = asd + buf * bufstride;
        float* ws = wsd + buf * bufstride;
        *(float4*)(as)     = a0;
        *(float4*)(as + 4) = a1;
        *(float4*)(ws)     = w0;
        *(float4*)(ws + 4) = w1;
    };

    v8f acc0 = {}, acc1 = {}, acc2 = {}, acc3 = {};
    const float* const Afrag = &As[0][(wave * 16 + lm) * LDS_S + khalf];
    const float* const Wf0   = &Ws[0][( 0 + lm) * LDS_S + khalf];
    const float* const Wf1   = &Ws[0][(16 + lm) * LDS_S + khalf];
    const float* const Wf2   = &Ws[0][(32 + lm) * LDS_S + khalf];
    const float* const Wf3   = &Ws[0][(48 + lm) * LDS_S + khalf];

    stage(0, 0);
    __syncthreads();

    for (int k0 = 0; k0 < K; k0 += TK_) {
        const int buf = (k0 / TK_) & 1;
        if (k0 + TK_ < K) stage(buf ^ 1, k0 + TK_);   // overlap next chunk's loads

        const int bo = buf * bufstride;
#pragma unroll
        for (int kk = 0; kk < TK_; kk += 4) {
            v2f a, b;
            a.x = Afrag[bo + kk];  a.y = Afrag[bo + kk + 1];
            b.x = Wf0[bo + kk];    b.y = Wf0[bo + kk + 1];
            acc0 = __builtin_amdgcn_wmma_f32_16x16x4_f32(false, a, false, b, (short)0, acc0, false, false);
            b.x = Wf1[bo + kk];    b.y = Wf1[bo + kk + 1];
            acc1 = __builtin_amdgcn_wmma_f32_16x16x4_f32(false, a, false, b, (short)0, acc1, false, false);
            b.x = Wf2[bo + kk];    b.y = Wf2[bo + kk + 1];
            acc2 = __builtin_amdgcn_wmma_f32_16x16x4_f32(false, a, false, b, (short)0, acc2, false, false);
            b.x = Wf3[bo + kk];    b.y = Wf3[bo + kk + 1];
            acc3 = __builtin_amdgcn_wmma_f32_16x16x4_f32(false, a, false, b, (short)0, acc3, false, false);
        }
        __syncthreads();   // reads of buf done; writes of buf^1 visible next iter
    }

    const int rbase = mt * 64 + wave * 16 + ((lane >= 16) ? 8 : 0);
    v8f accs[4] = {acc0, acc1, acc2, acc3};
#pragma unroll
    for (int j = 0; j < 4; ++j) {
        const int ncol = nt * 64 + (j << 4) + lm;
        const float bv = bias ? bias[ncol] : 0.0f;
#pragma unroll
        for (int v = 0; v < 8; ++v)
            D[(size_t)(rbase + v) * N + ncol] = accs[j][v] + bv;
    }
}

// ---------------------------------------------------------------------------
// Stable descending argsort of group_count (B=128), O(B^2) rank per thread.
// Also zeroes the loss accumulators.
// ---------------------------------------------------------------------------
__global__ __launch_bounds__(128) void sort_prep(
    const int* __restrict__ gc, int* __restrict__ idx_sorted,
    int* __restrict__ gc_sorted, float* __restrict__ accs)
{
    const int i  = threadIdx.x;
    const int my = gc[i];
    int rank = 0;
    for (int j = 0; j < B_; ++j) {
        const int gj = gc[j];
        if (gj > my || (gj == my && j < i)) ++rank;
    }
    idx_sorted[rank] = i;
    gc_sorted[rank]  = my;
    if (i < 2) accs[i] = 0.0f;
}

// ---------------------------------------------------------------------------
// inps[b,s,:] = sum_l alpha[b,s,l] * enc[b,l,:]   (enc NOT permuted, matching
// the reference; segments ARE permuted via idx_sorted). s==0 row is zero.
// ---------------------------------------------------------------------------
__global__ __launch_bounds__(256) void compute_inps(
    const float* __restrict__ enc, const int* __restrict__ segments,
    const int* __restrict__ idx_sorted, float* __restrict__ inps)
{
    const int b = blockIdx.x;   // sorted position == enc batch row
    const int s = blockIdx.y;
    const int t = threadIdx.x;
    float* __restrict__ out = inps + ((size_t)b * S_ + s) * E2_;

    if (s == 0) {
        for (int e = t; e < E2_; e += 256) out[e] = 0.0f;
        return;
    }

    __shared__ float segw[L_];
    __shared__ float inv_denom;
    const int src_b = idx_sorted[b];
    if (t < L_)
        segw[t] = (float)segments[((size_t)src_b * S_ + (s - 1)) * L_ + t];
    __syncthreads();
    if (t == 0) {
        float sum = 0.0f;
        for (int l = 0; l < L_; ++l) sum += segw[l];
        inv_denom = 1.0f / (sum + 1.0f);
    }
    __syncthreads();

    float acc[4] = {0.f, 0.f, 0.f, 0.f};
    const float* __restrict__ e0 = enc + (size_t)b * L_ * E2_;
    for (int l = 0; l < L_; ++l) {
        const float w = segw[l];
        if (w != 0.0f) {
            const float* __restrict__ er = e0 + (size_t)l * E2_;
#pragma unroll
            for (int q = 0; q < 4; ++q) acc[q] += w * er[t + q * 256];
        }
    }
    const float d = inv_denom;
#pragma unroll
    for (int q = 0; q < 4; ++q) out[t + q * 256] = acc[q] * d;
}

// ---------------------------------------------------------------------------
// GRU pointwise update for one step s. grid (B, 2), block 256 -> j in [0,512)
// ---------------------------------------------------------------------------
__global__ __launch_bounds__(256) void gru_pointwise(
    const float* __restrict__ xp, const float* __restrict__ gh,
    float* __restrict__ h, float* __restrict__ outs, int s)
{
    const int b = blockIdx.x;
    const int j = threadIdx.x + (blockIdx.y << 8);
    const float* __restrict__ x = xp + ((size_t)b * S_ + s) * (3 * H_);
    const float* __restrict__ g = gh + (size_t)b * (3 * H_);
    const float ir = x[j], iz = x[H_ + j], inn = x[2 * H_ + j];
    const float hr = g[j], hz = g[H_ + j], hn  = g[2 * H_ + j];
    const float r = 1.0f / (1.0f + expf(-(ir + hr)));
    const float z = 1.0f / (1.0f + expf(-(iz + hz)));
    const float n = tanhf(inn + r * hn);
    const float hv = h[(size_t)b * H_ + j];
    const float hnew = (1.0f - z) * n + z * hv;
    h[(size_t)b * H_ + j] = hnew;
    outs[((size_t)b * S_ + s) * H_ + j] = hnew;
}

__device__ __forceinline__ float softplusf(float x) {
    return fmaxf(x, 0.0f) + log1pf(expf(-fabsf(x)));
}

// ---------------------------------------------------------------------------
// Group (plan) BCE: for each (b,l), loop s < gc[b]:
//   logit = dot_e(tanh(hpart[b,s,e] + epart[b,l,e]), W_p2) + b_p2
//   acc  += softplus(logit) - logit * seg_sorted[b,s,l]
// ---------------------------------------------------------------------------
__global__ __launch_bounds__(256) void group_loss_kernel(
    const float* __restrict__ hpart, const float* __restrict__ epart,
    const float* __restrict__ W_p2, const float* __restrict__ b_p2,
    const int* __restrict__ segments, const int* __restrict__ idx_sorted,
    const int* __restrict__ gc_sorted, float* __restrict__ accs)
{
    const int b = blockIdx.x, l = blockIdx.y, t = threadIdx.x;
    const float ep0 = epart[((size_t)b * L_ + l) * H_ + t];
    const float ep1 = epart[((size_t)b * L_ + l) * H_ + t + 256];
    const float w0 = W_p2[t], w1 = W_p2[t + 256];
    const int src_b = idx_sorted[b];
    const int gcb   = gc_sorted[b];
    const float bp2 = b_p2[0];
    __shared__ float red[256];
    float local = 0.0f;
    for (int s = 0; s < gcb; ++s) {
        const float* __restrict__ hp = hpart + ((size_t)b * S_ + s) * H_;
        red[t] = tanhf(hp[t] + ep0) * w0 + tanhf(hp[t + 256] + ep1) * w1;
        __syncthreads();
        for (int st = 128; st > 0; st >>= 1) {
            if (t < st) red[t] += red[t + st];
            __syncthreads();
        }
        if (t == 0) {
            const float logit  = red[0] + bp2;
            const float target = (float)segments[((size_t)src_b * S_ + s) * L_ + l];
            local += softplusf(logit) - logit * target;
        }
        __syncthreads();
    }
    if (t == 0) atomicAdd(&accs[1], local);
}

// ---------------------------------------------------------------------------
// Stop BCE: per b, loop s < gc[b]: logit = dot(outputs[b,s,:], W_stop)+b_stop
// ---------------------------------------------------------------------------
__global__ __launch_bounds__(256) void stop_loss_kernel(
    const float* __restrict__ outs, const float* __restrict__ W_stop,
    const float* __restrict__ b_stop, const int* __restrict__ gc_sorted,
    float* __restrict__ accs)
{
    const int b = blockIdx.x, t = threadIdx.x;
    const float w0 = W_stop[t], w1 = W_stop[t + 256];
    const int gcb = gc_sorted[b];
    const float bs = b_stop[0];
    __shared__ float red[256];
    float local = 0.0f;
    for (int s = 0; s < gcb; ++s) {
        const float* __restrict__ o = outs + ((size_t)b * S_ + s) * H_;
        red[t] = o[t] * w0 + o[t + 256] * w1;
        __syncthreads();
        for (int st = 128; st > 0; st >>= 1) {
            if (t < st) red[t] += red[t + st];
            __syncthreads();
        }
        if (t == 0) {
            const float logit = red[0] + bs;
            const float label = (s == gcb - 1) ? 1.0f : 0.0f;
            local += softplusf(logit) - logit * label;
        }
        __syncthreads();
    }
    if (t == 0) atomicAdd(&accs[0], local);
}

__global__ void finalize_kernel(const int* __restrict__ gc_sorted,
                                const float* __restrict__ accs,
                                float* __restrict__ out)
{
    if (threadIdx.x == 0) {
        float msum = 0.0f;
        for (int b = 0; b < B_; ++b) msum += (float)gc_sorted[b];
        out[0] = accs[0] / msum;                     // stop_loss
        out[1] = accs[1] / (msum * (float)L_);       // group_loss
    }
}

// ---------------------------------------------------------------------------
extern "C" void kernel_launch(void* const* d_in, const int* in_sizes, int n_in,
                              void* d_out, int out_size, void* d_ws, size_t ws_size,
                              hipStream_t stream)
{
    const float* enc     = (const float*)d_in[0];   // (B,L,E2)
    const int*   seg     = (const int*)  d_in[1];   // (B,S,L)
    const int*   gcnt    = (const int*)  d_in[2];   // (B,)
    const float* gis     = (const float*)d_in[3];   // (B,2304)
    const float* W_init  = (const float*)d_in[4];   // (512,2304)
    const float* b_init  = (const float*)d_in[5];
    const float* W_ih    = (const float*)d_in[6];   // (1536,1024)
    const float* b_ih    = (const float*)d_in[7];
    const float* W_hh    = (const float*)d_in[8];   // (1536,512)
    const float* b_hh    = (const float*)d_in[9];
    const float* W_p1    = (const float*)d_in[10];  // (512,1536)
    const float* b_p1    = (const float*)d_in[11];
    const float* W_p2    = (const float*)d_in[12];  // (1,512)
    const float* b_p2    = (const float*)d_in[13];
    const float* W_stop  = (const float*)d_in[14];  // (1,512)
    const float* b_stop  = (const float*)d_in[15];
    float* out = (float*)d_out;

    // workspace layout (256B-aligned slices)
    char* ws = (char*)d_ws;
    size_t off = 0;
    auto alloc = [&](size_t bytes) { void* p = ws + off; off = (off + bytes + 255) & ~(size_t)255; return p; };
    float* inps    = (float*)alloc((size_t)B_ * S_ * E2_ * 4);       // 6.3 MB
    float* x_proj  = (float*)alloc((size_t)B_ * S_ * 3 * H_ * 4);    // 9.4 MB
    float* h       = (float*)alloc((size_t)B_ * H_ * 4);
    float* gh      = (float*)alloc((size_t)B_ * 3 * H_ * 4);
    float* outputs = (float*)alloc((size_t)B_ * S_ * H_ * 4);        // 3.1 MB
    float* hpart   = (float*)alloc((size_t)B_ * S_ * H_ * 4);        // 3.1 MB
    float* epart   = (float*)alloc((size_t)B_ * L_ * H_ * 4);        // 33.6 MB
    float* accs    = (float*)alloc(2 * 4);
    int*   idx_s   = (int*)  alloc(B_ * 4);
    int*   gc_s    = (int*)  alloc(B_ * 4);
    (void)ws_size; (void)in_sizes; (void)n_in; (void)out_size;

    const int init_in = 2304;

    // 1. stable descending argsort of group_count + zero accumulators
    sort_prep<<<1, 128, 0, stream>>>(gcnt, idx_s, gc_s, accs);

    // 2. inps = alpha @ enc (sorted segs, unsorted enc — matches reference)
    compute_inps<<<dim3(B_, S_), 256, 0, stream>>>(enc, seg, idx_s, inps);

    // 3. h0 = group_init_state @ W_init^T + b_init        (128x512, K=2304)
    wmma_gemm_nt<<<dim3(B_ / 64, H_ / 64), 128, 0, stream>>>(
        gis, W_init, b_init, h, B_, H_, init_in, init_in);

    // 4. x_proj = inps @ W_ih^T + b_ih                    (1536x1536, K=1024)
    wmma_gemm_nt<<<dim3(B_ * S_ / 64, 3 * H_ / 64), 128, 0, stream>>>(
        inps, W_ih, b_ih, x_proj, B_ * S_, 3 * H_, E2_, E2_);

    // 5. GRU scan: 12 sequential (WMMA gemm + pointwise) steps
    for (int s = 0; s < S_; ++s) {
        wmma_gemm_nt<<<dim3(B_ / 64, 3 * H_ / 64), 128, 0, stream>>>(
            h, W_hh, b_hh, gh, B_, 3 * H_, H_, H_);
        gru_pointwise<<<dim3(B_, 2), 256, 0, stream>>>(x_proj, gh, h, outputs, s);
    }

    // 6. hpart = outputs @ W1h^T   (W1h = W_p1[:, :512], row stride 1536)
    wmma_gemm_nt<<<dim3(B_ * S_ / 64, H_ / 64), 128, 0, stream>>>(
        outputs, W_p1, nullptr, hpart, B_ * S_, H_, H_, 3 * H_);

    // 7. epart = enc @ W1e^T + b_p1  (W1e = W_p1[:, 512:], 16384x512, K=1024)
    wmma_gemm_nt<<<dim3(B_ * L_ / 64, H_ / 64), 128, 0, stream>>>(
        enc, W_p1 + H_, b_p1, epart, B_ * L_, H_, E2_, 3 * H_);

    // 8. masked BCE reductions
    group_loss_kernel<<<dim3(B_, L_), 256, 0, stream>>>(
        hpart, epart, W_p2, b_p2, seg, idx_s, gc_s, accs);
    stop_loss_kernel<<<B_, 256, 0, stream>>>(outputs, W_stop, b_stop, gc_s, accs);

    // 9. normalize into d_out = {stop_loss, group_loss}
    finalize_kernel<<<1, 32, 0, stream>>>(gc_s, accs, out);
}